// NCEAverage_53455162966647
// MI455X (gfx1250) — compile-verified
//
#include <hip/hip_runtime.h>
#include <math.h>

#define BB    256
#define DD    128
#define NBANK 1000000
#define KK1   4096
#define INV_T (1.0f / 0.07f)
#define MOM   0.5f

typedef __attribute__((ext_vector_type(2))) float v2f;
typedef __attribute__((ext_vector_type(8))) float v8f;

// ---------------------------------------------------------------------------
// Kernel 1: gather + batched dot via V_WMMA_F32_16X16X4_F32, exp, partial sums
// One wave32 per (b, 16-wide k tile). D=128 -> 32 WMMA steps of K=4.
//   A (16x4 f32): lane m in [0,16) = gathered row (ktile*16+m); lanes 16..31
//                 carry the same rows' K=2,3 pair (ISA A layout).
//   B (4x16 f32): anchor[b] broadcast to all 16 columns (wave-uniform),
//                 same K-pair-per-lane-half layout as A.
//   D column 0 lives in lanes 0 (M=0..7) and 16 (M=8..15), one value per VGPR.
// ---------------------------------------------------------------------------
__global__ __launch_bounds__(256)
void nce_ev_wmma_kernel(const float* __restrict__ anchor,
                        const float* __restrict__ memory,
                        const int*   __restrict__ idx,
                        float*       __restrict__ ev_out,
                        float*       __restrict__ partials)
{
    const int tid      = threadIdx.x;
    const int lane     = tid & 31;
    const int wave     = tid >> 5;
    const int waveTile = blockIdx.x * 8 + wave;   // 65536 tiles total
    const int b        = waveTile >> 8;           // [0,256)
    const int ktile    = waveTile & 255;          // [0,256)
    const int half     = lane >> 4;               // 0 -> K={0,1}, 1 -> K={2,3}
    const int m        = lane & 15;               // row within tile

    const int    krow   = ktile * 16 + m;
    const int    rowIdx = idx[b * KK1 + krow];
    const float* arow   = memory + (size_t)rowIdx * DD + half * 2;
    const float* brow   = anchor + b * DD + half * 2;

    v8f c = {};
#pragma unroll 4
    for (int s = 0; s < 32; ++s) {
        v2f a  = *(const v2f*)(arow + s * 4);
        v2f bv = *(const v2f*)(brow + s * 4);
        // (neg_a, A, neg_b, B, c_mod, C, reuse_a, reuse_b)
        c = __builtin_amdgcn_wmma_f32_16x16x4_f32(false, a, false, bv,
                                                  (short)0, c, false, false);
    }

    float lsum = 0.0f;
    if (m == 0) {
        const int kbase = ktile * 16 + half * 8;
        float* dst = ev_out + (size_t)b * KK1 + kbase;
#pragma unroll
        for (int r = 0; r < 8; ++r) {
            float e = __expf(c[r] * INV_T);
            dst[r] = e;
            lsum  += e;
        }
    }

    // Deterministic block partial sum (no float atomics -> replay-stable).
    __shared__ float sred[256];
    sred[tid] = lsum;
    __syncthreads();
    for (int off = 128; off > 0; off >>= 1) {
        if (tid < off) sred[tid] += sred[tid + off];
        __syncthreads();
    }
    if (tid == 0) partials[blockIdx.x] = sred[0];
}

// ---------------------------------------------------------------------------
// Kernel 2: reduce 8192 block partials -> invZ = (B*K1) / (N * sum)
// Single block, fixed traversal order -> deterministic.
// ---------------------------------------------------------------------------
__global__ __launch_bounds__(256)
void nce_reduce_kernel(const float* __restrict__ partials,
                       float*       __restrict__ invZ,
                       int n)
{
    __shared__ float s[256];
    float acc = 0.0f;
    for (int i = threadIdx.x; i < n; i += 256) acc += partials[i];
    s[threadIdx.x] = acc;
    __syncthreads();
    for (int off = 128; off > 0; off >>= 1) {
        if (threadIdx.x < off) s[threadIdx.x] += s[threadIdx.x + off];
        __syncthreads();
    }
    if (threadIdx.x == 0) {
        float sum = s[0];
        // Z = mean(ev) * N = sum/(B*K1) * N ; out = ev * (B*K1)/(N*sum)
        invZ[0] = 1048576.0f / (1000000.0f * sum);
    }
}

// ---------------------------------------------------------------------------
// Kernel 3: out = ev * invZ (in place on d_out; recomputed fully each replay)
// ---------------------------------------------------------------------------
__global__ __launch_bounds__(256)
void nce_scale_kernel(float* __restrict__ out, const float* __restrict__ invZ)
{
    const int i = blockIdx.x * 256 + threadIdx.x;
    out[i] *= invZ[0];
}

// ---------------------------------------------------------------------------
// Kernel 4: momentum update + L2 norm + scatter (last-writer-wins for
// duplicate indices so replays are deterministic).
// ---------------------------------------------------------------------------
__global__ __launch_bounds__(128)
void nce_update_kernel(const float* __restrict__ anchor,
                       const float* __restrict__ memory,
                       const int*   __restrict__ index,
                       float*       __restrict__ newmem)
{
    const int b   = blockIdx.x;
    const int d   = threadIdx.x;
    const int row = index[b];
    for (int b2 = b + 1; b2 < BB; ++b2)
        if (index[b2] == row) return;   // a later b writes this row: yield

    float f = MOM * memory[(size_t)row * DD + d] + (1.0f - MOM) * anchor[b * DD + d];

    __shared__ float s[128];
    s[d] = f * f;
    __syncthreads();
    for (int off = 64; off > 0; off >>= 1) {
        if (d < off) s[d] += s[d + off];
        __syncthreads();
    }
    newmem[(size_t)row * DD + d] = f / sqrtf(s[0]);
}

// ---------------------------------------------------------------------------
extern "C" void kernel_launch(void* const* d_in, const int* in_sizes, int n_in,
                              void* d_out, int out_size, void* d_ws, size_t ws_size,
                              hipStream_t stream)
{
    const float* anchor = (const float*)d_in[0];   // [256,128]
    const float* memory = (const float*)d_in[1];   // [1e6,128]
    const int*   idx    = (const int*)  d_in[2];   // [256,4096]
    const int*   index  = (const int*)  d_in[3];   // [256]

    float* out_ev   = (float*)d_out;                          // [B*K1]
    float* newmem   = (float*)d_out + (size_t)BB * KK1;       // [N*D]
    float* partials = (float*)d_ws;                           // [8192]
    float* invZ     = partials + 8192;                        // [1]

    // 1) new_memory = memory (bulk 512 MB D2D)
    hipMemcpyAsync(newmem, memory, (size_t)NBANK * DD * sizeof(float),
                   hipMemcpyDeviceToDevice, stream);

    // 2) gathered WMMA dots -> ev + block partial sums (512 MB gather)
    nce_ev_wmma_kernel<<<8192, 256, 0, stream>>>(anchor, memory, idx,
                                                 out_ev, partials);

    // 3) global sum -> invZ
    nce_reduce_kernel<<<1, 256, 0, stream>>>(partials, invZ, 8192);

    // 4) out = ev * invZ
    nce_scale_kernel<<<(BB * KK1) / 256, 256, 0, stream>>>(out_ev, invZ);

    // 5) scatter updated rows into the bank copy
    nce_update_kernel<<<BB, 128, 0, stream>>>(anchor, memory, index, newmem);
}